// Model5_54185307406494
// MI455X (gfx1250) — compile-verified
//
#include <hip/hip_runtime.h>

typedef float v2f __attribute__((ext_vector_type(2)));
typedef float v8f __attribute__((ext_vector_type(8)));

constexpr int BN = 4, NN = 512, FIN = 32, HH = 64, TOUT = 12;
#define PI_F 3.14159265358979323846f

// ---- CDNA5 async global->LDS DMA (ASYNCcnt-tracked), per CDNA5 ISA ----
// LDS byte offset = low 32 bits of generic pointer.
__device__ __forceinline__ void async_ld_b128(void* lds_dst, const void* gsrc) {
  unsigned off = (unsigned)(unsigned long long)lds_dst;
  asm volatile("global_load_async_to_lds_b128 %0, %1, off"
               :
               : "v"(off), "v"(gsrc)
               : "memory");
}
__device__ __forceinline__ void wait_asynccnt0() {
  asm volatile("s_wait_asynccnt 0x0" ::: "memory");
}

// ---------------- downsample (avg-pool over T by 2) ----------------
__global__ void k_downsample(const float* __restrict__ in, float* __restrict__ out,
                             int Tout, int total) {
  int o = blockIdx.x * blockDim.x + threadIdx.x;
  if (o >= total) return;
  int f = o & 31;
  int t = (o >> 5) % Tout;
  int bn = (o >> 5) / Tout;
  const float* p = in + ((size_t)bn * (2 * Tout) + 2 * t) * 32 + f;
  out[o] = 0.5f * (p[0] + p[32]);
}

// ---------------- h = x @ W_gcn  -> layout [B*T, N, H] ----------------
__global__ void k_compute_h(const float* __restrict__ x, const float* __restrict__ W,
                            float* __restrict__ hbuf, int T) {
  __shared__ float sw[FIN * HH];
  for (int k = threadIdx.x; k < FIN * HH; k += blockDim.x) sw[k] = W[k];
  __syncthreads();
  int o = blockIdx.x * blockDim.x + threadIdx.x;
  int hh = o & 63;
  int n  = (o >> 6) & 511;
  int bt = o >> 15;                       // N*H = 32768 = 2^15
  int b = bt / T, t = bt - b * T;
  const float* xp = x + (((size_t)b * NN + n) * T + t) * FIN;
  float acc = 0.f;
#pragma unroll
  for (int f = 0; f < FIN; ++f) acc += xp[f] * sw[f * HH + hh];
  hbuf[o] = acc;
}

// ---------------- adaptive GCN: scores->relu->softmax->adj@h (+b, relu) ----------------
__global__ __launch_bounds__(128) void k_gcn(const float* __restrict__ x,
                                             const float* __restrict__ hbuf,
                                             const float* __restrict__ bias,
                                             float* __restrict__ enc, int T) {
  __shared__ float sXr[16][36];       // 16 rows x 32 feats (padded, 144B rows)
  __shared__ float sStage[64][68];    // staging: Xc 64x32 / Hc 64x64 (padded, 272B rows)
  __shared__ float sScore[16][516];   // score row-tile 16x512 (padded)

  const int tid  = threadIdx.x;
  const int lane = tid & 31;
  const int wave = tid >> 5;
  const int bt   = blockIdx.x;
  const int b    = bt / T, t = bt - b * T;
  const int r0   = blockIdx.y * 16;

  // async-DMA the 16-row tile of X_t straight into LDS
  {
    int row = tid >> 3, q = tid & 7;
    async_ld_b128(&sXr[row][q * 4],
                  x + (((size_t)b * NN + r0 + row) * T + t) * FIN + q * 4);
  }

  const int m16  = lane & 15;
  const int half = lane >> 4;

  // ---- phase 1: scores = Xr . Xc^T (WMMA f32 16x16x4, K=32 in 8 steps) ----
  for (int cb = 0; cb < NN / 64; ++cb) {
    __syncthreads();                      // previous chunk consumers done
#pragma unroll
    for (int r = 0; r < 4; ++r) {
      int idx = tid + r * 128;
      int row = idx >> 3, q = idx & 7;
      async_ld_b128(&sStage[row][q * 4],
                    x + (((size_t)b * NN + cb * 64 + row) * T + t) * FIN + q * 4);
    }
    wait_asynccnt0();                     // this wave's DMAs have landed in LDS
    __syncthreads();                      // all waves' DMAs visible
    v8f acc = {};
#pragma unroll
    for (int kk = 0; kk < 8; ++kk) {
      v2f A, Bv;
      A[0]  = sXr[m16][kk * 4 + half * 2];
      A[1]  = sXr[m16][kk * 4 + half * 2 + 1];
      Bv[0] = sStage[wave * 16 + m16][kk * 4 + half * 2];
      Bv[1] = sStage[wave * 16 + m16][kk * 4 + half * 2 + 1];
      acc = __builtin_amdgcn_wmma_f32_16x16x4_f32(false, A, false, Bv, (short)0, acc, false, false);
    }
#pragma unroll
    for (int v = 0; v < 8; ++v) {
      float val = acc[v];
      sScore[v + half * 8][cb * 64 + wave * 16 + m16] = val > 0.f ? val : 0.f; // relu
    }
  }
  __syncthreads();

  // ---- softmax over each 512-wide row (8 threads per row) ----
  {
    int row = tid >> 3, l8 = tid & 7;
    float mx = 0.f;                        // all values >= 0 after relu
    for (int c = l8; c < NN; c += 8) mx = fmaxf(mx, sScore[row][c]);
    mx = fmaxf(mx, __shfl_xor(mx, 1, 8));
    mx = fmaxf(mx, __shfl_xor(mx, 2, 8));
    mx = fmaxf(mx, __shfl_xor(mx, 4, 8));
    float sum = 0.f;
    for (int c = l8; c < NN; c += 8) {
      float e = expf(sScore[row][c] - mx);
      sScore[row][c] = e;
      sum += e;
    }
    sum += __shfl_xor(sum, 1, 8);
    sum += __shfl_xor(sum, 2, 8);
    sum += __shfl_xor(sum, 4, 8);
    float inv = 1.f / sum;
    for (int c = l8; c < NN; c += 8) sScore[row][c] *= inv;
  }

  // ---- phase 2: out = adj[16,512] @ H_t[512,64] ----
  v8f oacc = {};
  for (int nb = 0; nb < NN / 64; ++nb) {
    __syncthreads();
#pragma unroll
    for (int r = 0; r < 8; ++r) {
      int idx = tid + r * 128;
      int row = idx >> 4, q = idx & 15;
      async_ld_b128(&sStage[row][q * 4],
                    hbuf + (((size_t)bt * NN + nb * 64 + row) * HH) + q * 4);
    }
    wait_asynccnt0();
    __syncthreads();
#pragma unroll
    for (int kk = 0; kk < 16; ++kk) {
      v2f A, Bv;
      A[0]  = sScore[m16][nb * 64 + kk * 4 + half * 2];
      A[1]  = sScore[m16][nb * 64 + kk * 4 + half * 2 + 1];
      Bv[0] = sStage[kk * 4 + half * 2][wave * 16 + m16];
      Bv[1] = sStage[kk * 4 + half * 2 + 1][wave * 16 + m16];
      oacc = __builtin_amdgcn_wmma_f32_16x16x4_f32(false, A, false, Bv, (short)0, oacc, false, false);
    }
  }

  // epilogue: + bias, relu, store [B,N,T,H]
  int hcol = wave * 16 + m16;
  float bv = bias[hcol];
#pragma unroll
  for (int v = 0; v < 8; ++v) {
    int m = v + half * 8;
    float val = oacc[v] + bv;
    val = val > 0.f ? val : 0.f;
    enc[(((size_t)b * NN + r0 + m) * T + t) * HH + hcol] = val;
  }
}

// ---------------- DFT decompose: season/trend with top-5 freq mask ----------------
template <int T>
__global__ void k_dft(const float* __restrict__ enc, float* __restrict__ season,
                      float* __restrict__ trend) {
  constexpr int TF = T / 2 + 1;
  __shared__ float cs[T], sn[T];
  if (threadIdx.x < T) {
    float ang = (2.0f * PI_F * (float)threadIdx.x) / (float)T;
    cs[threadIdx.x] = cosf(ang);
    sn[threadIdx.x] = sinf(ang);
  }
  __syncthreads();
  int idx = blockIdx.x * blockDim.x + threadIdx.x;
  int hh = idx & 63;
  int n  = (idx >> 6) & 511;
  int b  = idx >> 15;
  size_t base = (((size_t)b * NN + n) * T) * HH + hh;

  float xv[T];
#pragma unroll
  for (int t = 0; t < T; ++t) xv[t] = enc[base + (size_t)t * HH];

  float re[TF], im[TF], mg[TF];
#pragma unroll
  for (int f = 0; f < TF; ++f) {
    float r = 0.f, i = 0.f;
#pragma unroll
    for (int t = 0; t < T; ++t) {
      int k = (f * t) % T;
      r += xv[t] * cs[k];
      i -= xv[t] * sn[k];
    }
    re[f] = r; im[f] = i;
    mg[f] = sqrtf(r * r + i * i);
  }
  mg[0] = 0.f;   // DC magnitude zeroed before thresholding (as in reference)

  // 5th largest magnitude (duplicates counted; lexicographic (mag, idx) tiebreak)
  float curv = 3.4e38f; int curi = TF;
  float thresh = 0.f;
#pragma unroll
  for (int it = 0; it < 5; ++it) {
    float bvv = -1.f; int bi = -1;
#pragma unroll
    for (int f = 0; f < TF; ++f) {
      bool lt = (mg[f] < curv) || (mg[f] == curv && f < curi);
      bool gt = (mg[f] > bvv) || (mg[f] == bvv && f > bi);
      if (lt && gt) { bvv = mg[f]; bi = f; }
    }
    curv = bvv; curi = bi;
  }
  thresh = curv;

#pragma unroll
  for (int f = 0; f < TF; ++f) {
    bool keep = !(mg[f] < thresh);     // where(freq < thresh, 0, xf)
    if (!keep) { re[f] = 0.f; im[f] = 0.f; }
  }

#pragma unroll
  for (int t = 0; t < T; ++t) {
    float s = re[0];
#pragma unroll
    for (int f = 1; f < TF - 1; ++f) {
      int k = (f * t) % T;
      s += 2.f * (re[f] * cs[k] - im[f] * sn[k]);
    }
    {
      int k = ((TF - 1) * t) % T;      // Nyquist term (weight 1, T even)
      s += re[TF - 1] * cs[k] - im[TF - 1] * sn[k];
    }
    s *= (1.0f / (float)T);
    season[base + (size_t)t * HH] = s;
    trend[base + (size_t)t * HH]  = xv[t] - s;
  }
}

// ---------------- cross-scale MLP over T: tgt += W2 @ gelu(W1 @ src + b1) + b2 ----------------
template <int TS, int TT>
__global__ void k_cross(float* __restrict__ tgt, const float* __restrict__ src,
                        const float* __restrict__ W1, const float* __restrict__ b1,
                        const float* __restrict__ W2, const float* __restrict__ b2) {
  __shared__ float sW1[TT * TS], sW2[TT * TT], sb1[TT], sb2[TT];
  for (int k = threadIdx.x; k < TT * TS; k += blockDim.x) sW1[k] = W1[k];
  for (int k = threadIdx.x; k < TT * TT; k += blockDim.x) sW2[k] = W2[k];
  if (threadIdx.x < TT) { sb1[threadIdx.x] = b1[threadIdx.x]; sb2[threadIdx.x] = b2[threadIdx.x]; }
  __syncthreads();
  int idx = blockIdx.x * blockDim.x + threadIdx.x;
  int hh = idx & 63;
  int n  = (idx >> 6) & 511;
  int b  = idx >> 15;
  size_t bs  = (((size_t)b * NN + n) * TS) * HH + hh;
  size_t bt_ = (((size_t)b * NN + n) * TT) * HH + hh;
  float s[TS];
#pragma unroll
  for (int t = 0; t < TS; ++t) s[t] = src[bs + (size_t)t * HH];
  float tmp[TT];
#pragma unroll
  for (int t = 0; t < TT; ++t) {
    float a = sb1[t];
#pragma unroll
    for (int u = 0; u < TS; ++u) a += sW1[t * TS + u] * s[u];
    tmp[t] = 0.5f * a * (1.f + erff(a * 0.7071067811865476f));   // exact gelu
  }
#pragma unroll
  for (int t = 0; t < TT; ++t) {
    float a = sb2[t];
#pragma unroll
    for (int u = 0; u < TT; ++u) a += sW2[t * TT + u] * tmp[u];
    tgt[bt_ + (size_t)t * HH] += a;   // + target (buffer holds target)
  }
}

// ---------------- enc = season + trend ----------------
__global__ void k_add(float* __restrict__ dst, const float* __restrict__ a,
                      const float* __restrict__ c, int total) {
  int o = blockIdx.x * blockDim.x + threadIdx.x;
  if (o < total) dst[o] = a[o] + c[o];
}

// ---------------- predict head (fused 1x1 convs), accumulates into out ----------------
template <int T>
__global__ void k_predict(const float* __restrict__ enc, const float* __restrict__ w1,
                          const float* __restrict__ b1, const float* __restrict__ w2,
                          const float* __restrict__ b2, float* __restrict__ out, int init) {
  int idx = blockIdx.x * blockDim.x + threadIdx.x;   // (b*N + n)
  if (idx >= BN * NN) return;
  size_t base = (size_t)idx * T * HH;
  float w2l[HH];
  float w2s = 0.f;
#pragma unroll
  for (int h = 0; h < HH; ++h) { w2l[h] = w2[h]; w2s += w2l[h]; }
  float y[T];
#pragma unroll
  for (int t = 0; t < T; ++t) {
    float a = 0.f;
#pragma unroll
    for (int h = 0; h < HH; ++h) a += enc[base + (size_t)t * HH + h] * w2l[h];
    y[t] = a;
  }
  float bb = b2[0];
#pragma unroll
  for (int o = 0; o < TOUT; ++o) {
    float a = 0.f;
#pragma unroll
    for (int t = 0; t < T; ++t) a += y[t] * w1[o * T + t];
    float d = a + b1[o] * w2s + bb;
    if (init) out[(size_t)idx * TOUT + o] = d;
    else      out[(size_t)idx * TOUT + o] += d;
  }
}

extern "C" void kernel_launch(void* const* d_in, const int* in_sizes, int n_in,
                              void* d_out, int out_size, void* d_ws, size_t ws_size,
                              hipStream_t stream) {
  (void)in_sizes; (void)n_in; (void)out_size; (void)ws_size;
  const float* x_enc = (const float*)d_in[0];
  const float* Wg    = (const float*)d_in[1];
  const float* bg    = (const float*)d_in[2];
  float* ws  = (float*)d_ws;
  float* out = (float*)d_out;

  const size_t sz_x1 = (size_t)BN * NN * 24 * FIN;
  const size_t sz_x2 = (size_t)BN * NN * 12 * FIN;
  const size_t e0 = (size_t)BN * NN * 48 * HH;
  size_t es[3] = {e0, e0 / 2, e0 / 4};
  size_t off = 0;
  float* x1 = ws + off; off += sz_x1;
  float* x2 = ws + off; off += sz_x2;
  float* enc[3]; float* sea[3]; float* tre[3];
  for (int i = 0; i < 3; ++i) { enc[i] = ws + off; off += es[i]; }
  for (int i = 0; i < 3; ++i) { sea[i] = ws + off; off += es[i]; }
  for (int i = 0; i < 3; ++i) { tre[i] = ws + off; off += es[i]; }
  float* hb = ws + off; off += e0;

  const int Ts[3] = {48, 24, 12};
  const float* xs[3] = {x_enc, x1, x2};

  k_downsample<<<(int)((sz_x1 + 255) / 256), 256, 0, stream>>>(x_enc, x1, 24, (int)sz_x1);
  k_downsample<<<(int)((sz_x2 + 255) / 256), 256, 0, stream>>>(x1, x2, 12, (int)sz_x2);

  for (int s = 0; s < 3; ++s) {
    int T = Ts[s];
    int tot = BN * T * NN * HH;
    k_compute_h<<<tot / 256, 256, 0, stream>>>(xs[s], Wg + s * FIN * HH, hb, T);
    k_gcn<<<dim3(BN * T, NN / 16), 128, 0, stream>>>(xs[s], hb, bg + s * HH, enc[s], T);
  }

  const int NBNH = BN * NN * HH;   // 131072
  for (int e = 0; e < 2; ++e) {
    k_dft<48><<<NBNH / 256, 256, 0, stream>>>(enc[0], sea[0], tre[0]);
    k_dft<24><<<NBNH / 256, 256, 0, stream>>>(enc[1], sea[1], tre[1]);
    k_dft<12><<<NBNH / 256, 256, 0, stream>>>(enc[2], sea[2], tre[2]);
    const float* const* di = (const float* const*)d_in;
    // season bottom-up: s_mlps[e][0] 48->24, s_mlps[e][1] 24->12
    k_cross<48, 24><<<NBNH / 256, 256, 0, stream>>>(sea[1], sea[0],
        di[3 + e * 8 + 0], di[3 + e * 8 + 1], di[3 + e * 8 + 2], di[3 + e * 8 + 3]);
    k_cross<24, 12><<<NBNH / 256, 256, 0, stream>>>(sea[2], sea[1],
        di[3 + e * 8 + 4], di[3 + e * 8 + 5], di[3 + e * 8 + 6], di[3 + e * 8 + 7]);
    // trend top-down: t_mlps[e][1] 12->24, then t_mlps[e][0] 24->48
    k_cross<12, 24><<<NBNH / 256, 256, 0, stream>>>(tre[1], tre[2],
        di[19 + e * 8 + 4], di[19 + e * 8 + 5], di[19 + e * 8 + 6], di[19 + e * 8 + 7]);
    k_cross<24, 48><<<NBNH / 256, 256, 0, stream>>>(tre[0], tre[1],
        di[19 + e * 8 + 0], di[19 + e * 8 + 1], di[19 + e * 8 + 2], di[19 + e * 8 + 3]);
    for (int i = 0; i < 3; ++i)
      k_add<<<(int)(es[i] / 256), 256, 0, stream>>>(enc[i], sea[i], tre[i], (int)es[i]);
  }

  const float* const* di = (const float* const*)d_in;
  k_predict<48><<<(BN * NN) / 256, 256, 0, stream>>>(enc[0],
      di[35 + 0], di[35 + 1], di[35 + 2], di[35 + 3], out, 1);
  k_predict<24><<<(BN * NN) / 256, 256, 0, stream>>>(enc[1],
      di[35 + 4], di[35 + 5], di[35 + 6], di[35 + 7], out, 0);
  k_predict<12><<<(BN * NN) / 256, 256, 0, stream>>>(enc[2],
      di[35 + 8], di[35 + 9], di[35 + 10], di[35 + 11], out, 0);
}